// MLS_12326556139626
// MI455X (gfx1250) — compile-verified
//
#include <hip/hip_runtime.h>
#include <hip/hip_bf16.h>
#include <stdint.h>

#define TILE   64          // output tile is TILE x TILE per block
#define DCHUNK 16          // d-elements staged per chunk
#define LST    (DCHUNK + 4) // padded LDS row stride: 20 floats = 80B (16B aligned, bank-conflict free)

typedef int v4i __attribute__((vector_size(16)));
typedef __attribute__((address_space(1))) v4i gv4i_t;     // global int4
typedef __attribute__((address_space(3))) v4i lv4i_t;     // LDS int4
typedef __attribute__((address_space(1))) void gvoid_t;   // global (asm fallback)
typedef __attribute__((address_space(3))) void lvoid_t;   // LDS (asm fallback)

// ---- CDNA5 async global->LDS copy (ASYNCcnt-tracked), 16 bytes per lane ----
#if __has_builtin(__builtin_amdgcn_global_load_async_to_lds_b128)
#  define ASYNC_COPY16(gp, lp)                                                \
     __builtin_amdgcn_global_load_async_to_lds_b128((gv4i_t*)(gp),            \
                                                    (lv4i_t*)(lp), 0, 0)
#else
   // ISA 15.18.3 opcode 98: vdst = 32-bit LDS byte address, vaddr = 64-bit global address
#  define ASYNC_COPY16(gp, lp)                                                \
     asm volatile("global_load_async_to_lds_b128 %0, %1, off"                 \
                  :: "v"((unsigned)(size_t)(lvoid_t*)(lp)),                   \
                     "v"((gvoid_t*)(gp)) : "memory")
#endif

#if __has_builtin(__builtin_amdgcn_s_wait_asynccnt)
#  define WAIT_ASYNC0() __builtin_amdgcn_s_wait_asynccnt(0)
#else
#  define WAIT_ASYNC0() asm volatile("s_wait_asynccnt 0" ::: "memory")
#endif

// ---------------------------------------------------------------------------
// Kernel 1: row-normalize mu and sigma_sq = exp(log_sigma). 1 block per row.
// ---------------------------------------------------------------------------
__global__ void __launch_bounds__(256) MLS_prep_kernel(
    const float* __restrict__ muX, const float* __restrict__ lsig,
    float* __restrict__ mun, float* __restrict__ sig, int D) {
  const int row = blockIdx.x;
  const int d   = threadIdx.x;              // blockDim.x == D == 256
  const size_t base = (size_t)row * D;

  const float v = muX[base + d];
  float s = v * v;
  #pragma unroll
  for (int off = 16; off > 0; off >>= 1) s += __shfl_xor(s, off, 32);

  __shared__ float part[8];
  if ((threadIdx.x & 31) == 0) part[threadIdx.x >> 5] = s;
  __syncthreads();
  float tot = 0.0f;
  #pragma unroll
  for (int w = 0; w < 8; ++w) tot += part[w];

  const float inv = 1.0f / fmaxf(sqrtf(tot), 1e-12f);
  mun[base + d] = v * inv;
  sig[base + d] = __expf(lsig[base + d]);   // v_exp_f32 path
}

// ---------------------------------------------------------------------------
// Kernel 2: 64x64 output tile per block; double-buffered async LDS staging.
// Each thread owns a 4x4 register sub-tile (i = ty+16r, j = tx+16c).
// ---------------------------------------------------------------------------
__global__ void __launch_bounds__(256) MLS_pair_kernel(
    const float* __restrict__ mu, const float* __restrict__ sig,
    float* __restrict__ out, int N, int D) {
  __shared__ float sm[2][4][TILE * LST];    // [buf][muI,sigI,muJ,sigJ][row*LST+d]

  const int tid = threadIdx.x;
  const int tx  = tid & 15;                 // j sub-index
  const int ty  = tid >> 4;                 // i sub-index (0..15)
  const int i0  = blockIdx.y * TILE;
  const int j0  = blockIdx.x * TILE;

  // Loader mapping: thread -> (row 0..63, 16B quad 0..3) per array
  const int lrow = tid >> 2;
  const int lq   = (tid & 3) * 4;
  const int irow = min(i0 + lrow, N - 1);
  const int jrow = min(j0 + lrow, N - 1);
  const float* g0 = mu  + (size_t)irow * D + lq;
  const float* g1 = sig + (size_t)irow * D + lq;
  const float* g2 = mu  + (size_t)jrow * D + lq;
  const float* g3 = sig + (size_t)jrow * D + lq;
  const int lofs = lrow * LST + lq;

#define ISSUE_CHUNK(b, d0) do {                      \
    ASYNC_COPY16(g0 + (d0), &sm[b][0][lofs]);        \
    ASYNC_COPY16(g1 + (d0), &sm[b][1][lofs]);        \
    ASYNC_COPY16(g2 + (d0), &sm[b][2][lofs]);        \
    ASYNC_COPY16(g3 + (d0), &sm[b][3][lofs]);        \
  } while (0)

  float acc[4][4];
  #pragma unroll
  for (int r = 0; r < 4; ++r)
    #pragma unroll
    for (int q = 0; q < 4; ++q) acc[r][q] = 0.0f;

  ISSUE_CHUNK(0, 0);
  const int NC = D / DCHUNK;

  for (int c = 0; c < NC; ++c) {
    const int b = c & 1;
    WAIT_ASYNC0();          // my chunk-c async copies have landed in LDS
    __syncthreads();        // everyone's have; prior compute on buf b^1 done
    if (c + 1 < NC) ISSUE_CHUNK(b ^ 1, (c + 1) * DCHUNK);  // prefetch next

    #pragma unroll 4
    for (int d = 0; d < DCHUNK; ++d) {
      float mi[4], si[4], mj[4], sj[4];
      #pragma unroll
      for (int r = 0; r < 4; ++r) {
        mi[r] = sm[b][0][(ty + 16 * r) * LST + d];   // broadcast reads
        si[r] = sm[b][1][(ty + 16 * r) * LST + d];
        mj[r] = sm[b][2][(tx + 16 * r) * LST + d];   // conflict-free (stride 20)
        sj[r] = sm[b][3][(tx + 16 * r) * LST + d];
      }
      #pragma unroll
      for (int r = 0; r < 4; ++r)
        #pragma unroll
        for (int q = 0; q < 4; ++q) {
          const float ss = si[r] + sj[q];
          const float df = mi[r] - mj[q];
          const float rc = __builtin_amdgcn_rcpf(ss + 1e-10f);  // v_rcp_f32
          const float lg = __builtin_amdgcn_logf(ss);           // v_log_f32 (log2)
          acc[r][q] = fmaf(df * df, rc, acc[r][q]);
          acc[r][q] = fmaf(lg, 0.69314718055994530942f, acc[r][q]); // *ln2
        }
    }
    __syncthreads();        // buf b fully consumed before it is re-filled
  }

  #pragma unroll
  for (int r = 0; r < 4; ++r) {
    const int i = i0 + ty + 16 * r;
    if (i < N) {
      #pragma unroll
      for (int q = 0; q < 4; ++q) {
        const int j = j0 + tx + 16 * q;
        if (j < N) out[(size_t)i * N + j] = -acc[r][q];
      }
    }
  }
#undef ISSUE_CHUNK
}

// ---------------------------------------------------------------------------
extern "C" void kernel_launch(void* const* d_in, const int* in_sizes, int n_in,
                              void* d_out, int out_size, void* d_ws, size_t ws_size,
                              hipStream_t stream) {
  const float* muX  = (const float*)d_in[0];
  const float* lsig = (const float*)d_in[1];
  float* out = (float*)d_out;

  const int D = 256;
  const int N = in_sizes[0] / D;

  float* mun = (float*)d_ws;              // N*D floats
  float* sg  = mun + (size_t)N * D;       // N*D floats (2 MB total in d_ws)

  MLS_prep_kernel<<<N, D, 0, stream>>>(muX, lsig, mun, sg, D);

  dim3 grid((N + TILE - 1) / TILE, (N + TILE - 1) / TILE);
  MLS_pair_kernel<<<grid, 256, 0, stream>>>(mun, sg, out, N, D);

  (void)n_in; (void)out_size; (void)ws_size;
}